// ModelNew_44684839748054
// MI455X (gfx1250) — compile-verified
//
#include <hip/hip_runtime.h>

// Mamba selective scan, chunked 3-pass for MI455X (gfx1250, wave32).
// B=8, L=2048, D=1024, N=16.
//   pass1 (!FINAL): per-chunk local scan from h=0; stores chunk-end h and sum(delta)
//   pass2         : carry propagation across chunks per (batch, 16-channel tile)
//   pass3 (FINAL) : re-run chunk seeded with carried-in h; y via V_WMMA_F32_16X16X4_F32
//
// h tile (16 d x 16 n) lives in registers with per-lane mapping
//   reg r = 2j+p  holds element (d = lane&15, n = 4j + p + 2*(lane>>4))
// which is simultaneously the 16x4 A-operand layout (row=lane&15, k=p+2*(lane>>4))
// and the 4x16 B-operand layout (col=lane&15, k=p+2*(lane>>4)) of WMMA f32 16x16x4.
// Per timestep we issue 4 WMMAs with A = C_t masked into row (t&15) and B = h,
// accumulating y for 16 timesteps as acc[row=t_local][col=d]; that tile maps to
// directly coalesced global stores (reg r -> timestep gb+r+8*h1, lanes -> d),
// so no LDS transpose is needed. The D_vec*u skip term is folded into the
// accumulator with one fmaf per step (element (t&15, m) lives on this lane).

#define BATCH_  8
#define LEN_    2048
#define DMODEL_ 1024
#define NSTATE_ 16
#define NTILES_ (DMODEL_ / 16)

typedef float v2f __attribute__((ext_vector_type(2)));
typedef float v8f __attribute__((ext_vector_type(8)));

__device__ __forceinline__ float fast_exp2(float x) { return __builtin_amdgcn_exp2f(x); }

template <bool FINAL>
__global__ __launch_bounds__(32)
void mamba_scan_pass(const float* __restrict__ u,
                     const float* __restrict__ dl,
                     const float* __restrict__ Am,
                     const float* __restrict__ Bm,
                     const float* __restrict__ Cm,
                     const float* __restrict__ Dv,
                     float* __restrict__ y,
                     float* __restrict__ hbuf,    // !FINAL: hend (out); FINAL: hin (in, may be null)
                     float* __restrict__ dsumBuf, // !FINAL: out
                     int NC, int CHUNK)
{
  const int lane = threadIdx.x;
  const int m    = lane & 15;   // channel within tile / accumulator column
  const int h1   = lane >> 4;   // half-wave select
  const int c    = blockIdx.x;  // chunk
  const int tile = blockIdx.y;  // channel tile (16 d)
  const int b    = blockIdx.z;  // batch
  const int d0   = tile * 16;
  const int t0   = c * CHUNK;
  const float LOG2E = 1.4426950408889634f;

  // Pre-scaled A (fold log2(e) so the decay is a single v_exp_f32 on exp2 args)
  float A2[8];
#pragma unroll
  for (int r = 0; r < 8; ++r) {
    const int n = 4 * (r >> 1) + (r & 1) + 2 * h1;
    A2[r] = Am[(size_t)(d0 + m) * NSTATE_ + n] * LOG2E;
  }

  float h[8];
  const size_t slot = (((size_t)b * NC + c) * NTILES_ + tile) * 32 + lane;
  if (FINAL && hbuf != nullptr) {
    const float4* hp = (const float4*)(hbuf + slot * 8);
    const float4 a0 = hp[0], a1 = hp[1];
    h[0] = a0.x; h[1] = a0.y; h[2] = a0.z; h[3] = a0.w;
    h[4] = a1.x; h[5] = a1.y; h[6] = a1.z; h[7] = a1.w;
  } else {
#pragma unroll
    for (int r = 0; r < 8; ++r) h[r] = 0.f;
  }

  const float dD = FINAL ? Dv[d0 + m] : 0.f;
  // 0/1 masks selecting which half-wave owns accumulator rows 0-7 vs 8-15
  const float selLo = (h1 == 0) ? 1.f : 0.f;
  const float selHi = 1.f - selLo;
  float dsum = 0.f;
  v8f acc = {0.f, 0.f, 0.f, 0.f, 0.f, 0.f, 0.f, 0.f};

  const float* up = u  + ((size_t)b * LEN_ + t0) * DMODEL_ + d0 + m;
  const float* dp = dl + ((size_t)b * LEN_ + t0) * DMODEL_ + d0 + m;
  const float4* Bp = (const float4*)(Bm + ((size_t)b * LEN_ + t0) * NSTATE_);
  const float4* Cp = (const float4*)(Cm + ((size_t)b * LEN_ + t0) * NSTATE_);

  for (int gb = 0; gb < CHUNK; gb += 16) {
#pragma unroll
    for (int cc = 0; cc < 16; ++cc) {
      const int t = gb + cc;
      const float dt = dp[(size_t)t * DMODEL_];
      const float ut = up[(size_t)t * DMODEL_];
      // B_t: 16 wave-uniform floats (SMEM) -> per-reg operand via half-wave select
      const float4 B0 = Bp[t * 4 + 0], B1 = Bp[t * 4 + 1];
      const float4 B2 = Bp[t * 4 + 2], B3 = Bp[t * 4 + 3];
      float bv[8];
      bv[0] = h1 ? B0.z : B0.x;  bv[1] = h1 ? B0.w : B0.y;
      bv[2] = h1 ? B1.z : B1.x;  bv[3] = h1 ? B1.w : B1.y;
      bv[4] = h1 ? B2.z : B2.x;  bv[5] = h1 ? B2.w : B2.y;
      bv[6] = h1 ? B3.z : B3.x;  bv[7] = h1 ? B3.w : B3.y;
      const float du = dt * ut;
#pragma unroll
      for (int r = 0; r < 8; ++r) {
        const float dA = fast_exp2(dt * A2[r]);         // exp(delta*A)
        h[r] = fmaf(dA, h[r], du * bv[r]);              // h = dA*h + (d*u)*B
      }
      if (!FINAL) dsum += dt;
      if (FINAL) {
        const float4 C0 = Cp[t * 4 + 0], C1 = Cp[t * 4 + 1];
        const float4 C2 = Cp[t * 4 + 2], C3 = Cp[t * 4 + 3];
        float cv[8];
        cv[0] = h1 ? C0.z : C0.x;  cv[1] = h1 ? C0.w : C0.y;
        cv[2] = h1 ? C1.z : C1.x;  cv[3] = h1 ? C1.w : C1.y;
        cv[4] = h1 ? C2.z : C2.x;  cv[5] = h1 ? C2.w : C2.y;
        cv[6] = h1 ? C3.z : C3.x;  cv[7] = h1 ? C3.w : C3.y;
        const bool sel = (m == cc);  // C_t occupies only A-operand row cc
#pragma unroll
        for (int j = 0; j < 4; ++j) {
          v2f aop = {sel ? cv[2 * j] : 0.f, sel ? cv[2 * j + 1] : 0.f};
          v2f bop = {h[2 * j], h[2 * j + 1]};
          // acc[row=cc][col=d] += sum_k C_t[4j+k] * h[d, 4j+k]
          acc = __builtin_amdgcn_wmma_f32_16x16x4_f32(
              false, aop, false, bop, (short)0, acc, false, false);
        }
        // skip connection: y[t][m] += D[m]*u_t[m]; element (row cc, col m) lives
        // in reg cc&7 on the (cc>>3) half-wave, where this lane holds col==m.
        acc[cc & 7] = fmaf((cc < 8) ? selLo : selHi, dD * ut, acc[cc & 7]);
        if (cc == 15) {
          // acc reg r = timestep gb + r + 8*h1, lanes = 16 consecutive channels
          float* yp = y + ((size_t)b * LEN_ + t0 + gb + 8 * h1) * DMODEL_ + d0 + m;
#pragma unroll
          for (int r = 0; r < 8; ++r)
            yp[(size_t)r * DMODEL_] = acc[r];
#pragma unroll
          for (int r = 0; r < 8; ++r) acc[r] = 0.f;
        }
      }
    }
  }

  if (!FINAL) {
    float4* hp = (float4*)(hbuf + slot * 8);
    hp[0] = make_float4(h[0], h[1], h[2], h[3]);
    hp[1] = make_float4(h[4], h[5], h[6], h[7]);
    if (h1 == 0)
      dsumBuf[((size_t)b * NC + c) * DMODEL_ + d0 + m] = dsum;
  }
}

__global__ __launch_bounds__(32)
void mamba_carry_pass(const float* __restrict__ Am,
                      const float* __restrict__ hend,
                      float* __restrict__ hin,
                      const float* __restrict__ dsumBuf,
                      int NC)
{
  const int lane = threadIdx.x, m = lane & 15, h1 = lane >> 4;
  const int tile = blockIdx.x, b = blockIdx.y;
  const int d0 = tile * 16;
  const float LOG2E = 1.4426950408889634f;
  float A2[8];
#pragma unroll
  for (int r = 0; r < 8; ++r) {
    const int n = 4 * (r >> 1) + (r & 1) + 2 * h1;
    A2[r] = Am[(size_t)(d0 + m) * NSTATE_ + n] * LOG2E;
  }
  float carry[8];
#pragma unroll
  for (int r = 0; r < 8; ++r) carry[r] = 0.f;
  for (int c = 0; c < NC; ++c) {
    const size_t slot = (((size_t)b * NC + c) * NTILES_ + tile) * 32 + lane;
    float4* op = (float4*)(hin + slot * 8);               // state entering chunk c
    op[0] = make_float4(carry[0], carry[1], carry[2], carry[3]);
    op[1] = make_float4(carry[4], carry[5], carry[6], carry[7]);
    const float ds = dsumBuf[((size_t)b * NC + c) * DMODEL_ + d0 + m];
    const float4* ip = (const float4*)(hend + slot * 8);
    const float4 e0 = ip[0], e1 = ip[1];
    const float he[8] = {e0.x, e0.y, e0.z, e0.w, e1.x, e1.y, e1.z, e1.w};
#pragma unroll
    for (int r = 0; r < 8; ++r)                            // carry = exp(A*sum_d)*carry + hend
      carry[r] = fmaf(fast_exp2(A2[r] * ds), carry[r], he[r]);
  }
}

extern "C" void kernel_launch(void* const* d_in, const int* in_sizes, int n_in,
                              void* d_out, int out_size, void* d_ws, size_t ws_size,
                              hipStream_t stream) {
  (void)in_sizes; (void)n_in; (void)out_size;
  const float* u  = (const float*)d_in[0];
  const float* dl = (const float*)d_in[1];
  const float* Am = (const float*)d_in[2];
  const float* Bm = (const float*)d_in[3];
  const float* Cm = (const float*)d_in[4];
  const float* Dv = (const float*)d_in[5];
  float* y = (float*)d_out;

  auto need = [](int nc) -> size_t {
    const size_t hendE = (size_t)BATCH_ * nc * NTILES_ * 256;
    return (2 * hendE + (size_t)BATCH_ * nc * DMODEL_) * sizeof(float);
  };
  int NC = 16;
  while (NC > 1 && need(NC) > ws_size) NC >>= 1;
  const bool have_ws = (d_ws != nullptr) && (need(NC) <= ws_size);
  if (!have_ws) NC = 1;
  const int CHUNK = LEN_ / NC;

  float* ws = (float*)d_ws;
  const size_t hendE = (size_t)BATCH_ * NC * NTILES_ * 256;
  float* ws_hend = ws;
  float* ws_hin  = ws + hendE;
  float* ws_dsum = ws + 2 * hendE;

  const dim3 blk(32, 1, 1);
  if (have_ws) {
    if (NC > 1) {
      mamba_scan_pass<false><<<dim3(NC, NTILES_, BATCH_), blk, 0, stream>>>(
          u, dl, Am, Bm, Cm, Dv, y, ws_hend, ws_dsum, NC, CHUNK);
    }
    mamba_carry_pass<<<dim3(NTILES_, BATCH_, 1), blk, 0, stream>>>(
        Am, ws_hend, ws_hin, ws_dsum, NC);
    mamba_scan_pass<true><<<dim3(NC, NTILES_, BATCH_), blk, 0, stream>>>(
        u, dl, Am, Bm, Cm, Dv, y, ws_hin, nullptr, NC, CHUNK);
  } else {
    mamba_scan_pass<true><<<dim3(1, NTILES_, BATCH_), blk, 0, stream>>>(
        u, dl, Am, Bm, Cm, Dv, y, nullptr, nullptr, 1, LEN_);
  }
}